// SphereByCurvature_352187319088
// MI455X (gfx1250) — compile-verified
//
#include <hip/hip_runtime.h>
#include <hip/hip_bf16.h>
#include <math.h>

// ---------------- problem constants (from reference) ----------------
#define NUM_ITER 12
#define DAMPING  0.95f
#define TOLF     1e-6f
#define EPSF     1e-12f

// ---------------- tiling ----------------
#define TPB         256
#define TILE_RAYS   512
#define TILE_FLOATS (TILE_RAYS * 3)     // 1536 floats per tensor (P or V)
#define TILE_CHUNKS (TILE_FLOATS / 4)   // 384 float4 chunks per tensor
#define BUF_FLOATS  (TILE_FLOATS * 2)   // P tile then V tile

// ---------------- CDNA5 async global->LDS path (guarded) ----------------
typedef int v4i __attribute__((vector_size(4 * sizeof(int))));
#if defined(__AMDGCN__)
typedef __attribute__((address_space(1))) v4i* async_gptr_t;  // global src
typedef __attribute__((address_space(3))) v4i* async_lptr_t;  // LDS dst
#  if __has_builtin(__builtin_amdgcn_global_load_async_to_lds_b128)
#    define HAVE_ASYNC_LDS 1
#  endif
#  if __has_builtin(__builtin_amdgcn_s_wait_asynccnt)
#    define WAIT_ASYNCCNT(n) __builtin_amdgcn_s_wait_asynccnt(n)
#  else
#    define WAIT_ASYNCCNT(n) asm volatile("s_wait_asynccnt %0" ::"n"(n) : "memory")
#  endif
#endif
#ifndef HAVE_ASYNC_LDS
#  define HAVE_ASYNC_LDS 0
#endif
#ifndef WAIT_ASYNCCNT
#  define WAIT_ASYNCCNT(n) do {} while (0)
#endif

__device__ __forceinline__ float fast_rsq(float x) {
#if defined(__AMDGCN__)
  return __builtin_amdgcn_rsqf(x);   // v_rsq_f32 (trans unit, co-executes with VALU)
#else
  return 1.0f / sqrtf(x);
#endif
}
__device__ __forceinline__ float fast_rcp(float x) {
#if defined(__AMDGCN__)
  return __builtin_amdgcn_rcpf(x);   // v_rcp_f32
#else
  return 1.0f / x;
#endif
}

// ---------------- main kernel: async-staged, double-buffered Newton ----------------
// (Emitted first in the module so the disasm snippet shows the async staging.)
__global__ __launch_bounds__(TPB) void sphere_newton_kernel(
    const float* __restrict__ gP, const float* __restrict__ gV,
    const float* __restrict__ params,
    float* __restrict__ out_t, float* __restrict__ out_n,
    float* __restrict__ out_valid, long long N, int numTiles) {
  __shared__ float smem[2][BUF_FLOATS];   // 2 x 12 KB double buffer
  const int tid = threadIdx.x;

  // uniform parameters -> scalar loads
  float Ai[9];
#pragma unroll
  for (int i = 0; i < 9; ++i) Ai[i] = params[i];
  const float bx = params[9], by = params[10], bz = params[11];
  const float Cc = params[12];
  const float vR2 = params[13];
  const float C2 = Cc * Cc;

  const long long maxP = 3LL * N;   // floats per tensor

  auto stage = [&](int tile, int bufIdx) {
    const long long base3 = (long long)tile * TILE_FLOATS;
    float* buf = smem[bufIdx];
#pragma unroll
    for (int k = 0; k < 3; ++k) {
      const int chunk = tid + k * TPB;                 // 0..767, wave-uniform P/V split
      const int isV   = (chunk >= TILE_CHUNKS);
      const int c     = isV ? (chunk - TILE_CHUNKS) : chunk;
      long long g = base3 + (long long)c * 4;
      if (g + 4 > maxP) g = maxP - 4;                  // clamp tail tile (memory-safe)
      if (g < 0) g = 0;
      const float* gsrc = (isV ? gV : gP) + g;
      float* ldst = buf + (isV ? TILE_FLOATS : 0) + c * 4;
#if HAVE_ASYNC_LDS
      __builtin_amdgcn_global_load_async_to_lds_b128(
          (async_gptr_t)(gsrc), (async_lptr_t)(ldst), 0, 0);
#else
      const float4 v = *(const float4*)gsrc;
      *(float4*)ldst = v;
#endif
    }
  };

  int tile = blockIdx.x;
  int curBuf = 0;
  if (tile < numTiles) stage(tile, curBuf);            // prologue prefetch

  for (; tile < numTiles; tile += (int)gridDim.x) {
    const int nextTile = tile + (int)gridDim.x;
    if (nextTile < numTiles) {
      stage(nextTile, curBuf ^ 1);                     // prefetch next tile
#if HAVE_ASYNC_LDS
      WAIT_ASYNCCNT(3);                                // current tile's 3 ops landed
#endif
    } else {
#if HAVE_ASYNC_LDS
      WAIT_ASYNCCNT(0);
#endif
    }
    __syncthreads();                                   // all waves' staging visible

    const float* buf = smem[curBuf];
    const long long tileBase = (long long)tile * TILE_RAYS;

#pragma unroll
    for (int h = 0; h < 2; ++h) {
      const int l = tid + h * TPB;                     // conflict-free stride-3 LDS reads
      const long long ray = tileBase + l;
      const float px = buf[l * 3 + 0];
      const float py = buf[l * 3 + 1];
      const float pz = buf[l * 3 + 2];
      const float vx = buf[TILE_FLOATS + l * 3 + 0];
      const float vy = buf[TILE_FLOATS + l * 3 + 1];
      const float vz = buf[TILE_FLOATS + l * 3 + 2];

      // Pl = Ainv @ (P - b), Vl = Ainv @ V
      const float qx = px - bx, qy = py - by, qz = pz - bz;
      const float Plx = fmaf(Ai[0], qx, fmaf(Ai[1], qy, Ai[2] * qz));
      const float Ply = fmaf(Ai[3], qx, fmaf(Ai[4], qy, Ai[5] * qz));
      const float Plz = fmaf(Ai[6], qx, fmaf(Ai[7], qy, Ai[8] * qz));
      const float Vlx = fmaf(Ai[0], vx, fmaf(Ai[1], vy, Ai[2] * vz));
      const float Vly = fmaf(Ai[3], vx, fmaf(Ai[4], vy, Ai[5] * vz));
      const float Vlz = fmaf(Ai[6], vx, fmaf(Ai[7], vy, Ai[8] * vz));

      float t = 0.0f;
#pragma unroll
      for (int it = 0; it < NUM_ITER; ++it) {
        const float x = fmaf(t, Vlx, Plx);
        const float y = fmaf(t, Vly, Ply);
        const float z = fmaf(t, Vlz, Plz);
        const float r2 = fmaf(y, y, z * z);
        const float m = fmaxf(fmaf(-C2, r2, 1.0f), EPSF);
        const float inv_s = fast_rsq(m);
        const float s = m * inv_s;
        const float g = Cc * r2 * fast_rcp(1.0f + s);
        const float dgd = Cc * inv_s;
        const float f = g - x;
        const float denom = fmaf(dgd, fmaf(y, Vly, z * Vlz), -Vlx);
        // branch-free clamp: |denom| >= EPS, sign per (denom >= 0)
        const float mag = fmaxf(fabsf(denom), EPSF);
        const float dclamped = (denom >= 0.0f) ? mag : -mag;
        t = fmaf(-DAMPING * f, fast_rcp(dclamped), t);
      }

      // final point, gradient, world normal, validity
      const float y = fmaf(t, Vly, Ply);
      const float z = fmaf(t, Vlz, Plz);
      const float r2 = fmaf(y, y, z * z);
      const float m = fmaxf(fmaf(-C2, r2, 1.0f), EPSF);
      const float dgd = Cc * fast_rsq(m);
      const float gxc = -1.0f, gyc = dgd * y, gzc = dgd * z;
      // n_w = grad @ Ainv (row-vector times matrix)
      const float nwx = fmaf(gxc, Ai[0], fmaf(gyc, Ai[3], gzc * Ai[6]));
      const float nwy = fmaf(gxc, Ai[1], fmaf(gyc, Ai[4], gzc * Ai[7]));
      const float nwz = fmaf(gxc, Ai[2], fmaf(gyc, Ai[5], gzc * Ai[8]));
      const float inv_n = fast_rsq(fmaf(nwx, nwx, fmaf(nwy, nwy, nwz * nwz)));
      const float valid = (r2 <= vR2) ? 1.0f : 0.0f;

      if (ray < N) {
        __builtin_nontemporal_store(t, out_t + ray);
        __builtin_nontemporal_store(nwx * inv_n, out_n + ray * 3 + 0);
        __builtin_nontemporal_store(nwy * inv_n, out_n + ray * 3 + 1);
        __builtin_nontemporal_store(nwz * inv_n, out_n + ray * 3 + 2);
        __builtin_nontemporal_store(valid, out_valid + ray);
      }
    }
    __syncthreads();   // protect curBuf before it is restaged next iteration
    curBuf ^= 1;
  }
}

// ---------------- setup kernel: 4x4 algebra + A^-1 into workspace ----------------
// params layout (floats): [0..8] Ainv row-major, [9..11] b, [12] C, [13] (half+TOL)^2
__global__ void sphere_setup_kernel(const float* __restrict__ tf,
                                    const float* __restrict__ diameter,
                                    const float* __restrict__ Cin,
                                    const float* __restrict__ anchors,
                                    const float* __restrict__ scale,
                                    float* __restrict__ params,
                                    float* __restrict__ out_tfs,
                                    float* __restrict__ out_tfn) {
  if (threadIdx.x != 0 || blockIdx.x != 0) return;
  const float Cc   = Cin[0];
  const float half = diameter[0] * 0.5f;
  const float sc   = scale[0];

  auto sag_g = [&](float y) -> float {
    const float r2 = y * y;
    const float s  = sqrtf(fmaxf(1.0f - Cc * Cc * r2, EPSF));
    return Cc * r2 / (1.0f + s);
  };
  const float e0 = sag_g(anchors[0] * half);
  const float e1 = sag_g(anchors[1] * half);
  const float d0 = -sc * e0;            // tf @ translate_x(-scale*e0) @ S
  const float dn = sc * (e1 - e0);      // tf @ translate_x(scale*(e1-e0))

  float A[9], b3[3];
  for (int i = 0; i < 4; ++i) {
    for (int j = 0; j < 3; ++j) {
      const float v = tf[i * 4 + j];
      out_tfs[i * 4 + j] = v * sc;      // columns 0..2 scaled by S
      out_tfn[i * 4 + j] = v;
    }
    out_tfs[i * 4 + 3] = tf[i * 4 + 3] + tf[i * 4 + 0] * d0;
    out_tfn[i * 4 + 3] = tf[i * 4 + 3] + tf[i * 4 + 0] * dn;
  }
  for (int i = 0; i < 3; ++i) {
    for (int j = 0; j < 3; ++j) A[i * 3 + j] = out_tfs[i * 4 + j];
    b3[i] = out_tfs[i * 4 + 3];
  }
  // 3x3 inverse via adjugate
  const float c00 = A[4] * A[8] - A[5] * A[7];
  const float c01 = A[5] * A[6] - A[3] * A[8];
  const float c02 = A[3] * A[7] - A[4] * A[6];
  const float det = A[0] * c00 + A[1] * c01 + A[2] * c02;
  const float id  = 1.0f / det;
  params[0] = c00 * id;
  params[1] = (A[2] * A[7] - A[1] * A[8]) * id;
  params[2] = (A[1] * A[5] - A[2] * A[4]) * id;
  params[3] = c01 * id;
  params[4] = (A[0] * A[8] - A[2] * A[6]) * id;
  params[5] = (A[2] * A[3] - A[0] * A[5]) * id;
  params[6] = c02 * id;
  params[7] = (A[1] * A[6] - A[0] * A[7]) * id;
  params[8] = (A[0] * A[4] - A[1] * A[3]) * id;
  params[9]  = b3[0];
  params[10] = b3[1];
  params[11] = b3[2];
  params[12] = Cc;
  const float vr = half + TOLF;
  params[13] = vr * vr;
  // NOTE: tau cancels exactly in t-update and in the normalized normal -> dropped.
}

// ---------------- launch ----------------
extern "C" void kernel_launch(void* const* d_in, const int* in_sizes, int n_in,
                              void* d_out, int out_size, void* d_ws, size_t ws_size,
                              hipStream_t stream) {
  const float* P        = (const float*)d_in[0];
  const float* V        = (const float*)d_in[1];
  const float* tf       = (const float*)d_in[2];
  const float* diameter = (const float*)d_in[3];
  const float* C        = (const float*)d_in[4];
  const float* anchors  = (const float*)d_in[5];
  const float* scale    = (const float*)d_in[6];
  // d_in[7] = normalize: tau cancels analytically, unused.

  const long long N = (long long)(in_sizes[0] / 3);
  float* out       = (float*)d_out;
  float* out_t     = out;               // [0, N)
  float* out_n     = out + N;           // [N, 4N)
  float* out_valid = out + 4 * N;       // [4N, 5N)
  float* out_tfs   = out + 5 * N;       // 16 floats
  float* out_tfn   = out_tfs + 16;      // 16 floats
  float* params    = (float*)d_ws;

  sphere_setup_kernel<<<1, 1, 0, stream>>>(tf, diameter, C, anchors, scale,
                                           params, out_tfs, out_tfn);
  if (N <= 0) return;
  const long long numTiles = (N + TILE_RAYS - 1) / TILE_RAYS;
  const int grid = (int)(numTiles < 2048 ? numTiles : 2048);
  sphere_newton_kernel<<<grid, TPB, 0, stream>>>(P, V, params, out_t, out_n,
                                                 out_valid, N, (int)numTiles);
}